// HermiteFuncs_41721312314117
// MI455X (gfx1250) — compile-verified
//
#include <hip/hip_runtime.h>
#include <math.h>

typedef float v4f __attribute__((ext_vector_type(4)));

#define BLOCK 256
#define CHUNKS_PER_THREAD 8

// --- CDNA5 async global->LDS copy (ASYNCcnt path), 16 bytes per lane ---
__device__ __forceinline__ void async_load_f4_to_lds(unsigned lds_byte_addr,
                                                     const v4f* gaddr) {
    // VDST = LDS byte address (low 32 bits of the flat shared pointer are the
    // LDS offset per the aperture mapping), VADDR = 64-bit global addr pair.
    asm volatile("global_load_async_to_lds_b128 %0, %1, off"
                 :: "v"(lds_byte_addr), "v"(gaddr)
                 : "memory");
}

__device__ __forceinline__ void wait_async_le(int keep) {
    if (keep == 1)
        asm volatile("s_wait_asynccnt 0x1" ::: "memory");
    else
        asm volatile("s_wait_asynccnt 0x0" ::: "memory");
}

__device__ __forceinline__ float fast_tanh(float x) {
#if __has_builtin(__builtin_amdgcn_tanhf)
    return __builtin_amdgcn_tanhf(x);   // v_tanh_f32 (gfx1250 TRANS op)
#else
    float r;
    // v_nop after the TRANS op: ISA §7.4 requires 1 independent op before the
    // result is consumed; inline asm bypasses compiler hazard handling.
    asm("v_tanh_f32 %0, %1\n\tv_nop" : "=v"(r) : "v"(x));
    return r;
#endif
}

template <bool EXACT>
__global__ __launch_bounds__(BLOCK)
void HermiteFuncs_kernel(const v4f* __restrict__ x4,
                         v4f* __restrict__ out4,
                         int total4)  // number of float4 chunks in x
{
    __shared__ v4f buf[2][BLOCK];

    const int tid    = threadIdx.x;
    const int stride = gridDim.x * BLOCK;
    int q = blockIdx.x * BLOCK + tid;

    // Constants (R = 6, scale = sqrt(2R+1) = sqrt(13))
    const float S   = 3.6055512754639893f;      // sqrt(13)
    const float C0  = 0.7511255444649425f;      // pi^(-1/4)
    const float C1  = 1.0622519320271964f;      // sqrt(2) * pi^(-1/4)
    const float NE  = -0.7213475204444817f;     // -0.5 * log2(e)
    const float B2  = 0.7071067811865476f;      // sqrt(1/2)
    const float A3  = 0.8164965809277260f;      // sqrt(2/3)
    const float B3  = 0.8164965809277260f;      // sqrt(2/3)
    const float A4  = 0.7071067811865476f;      // sqrt(2/4)
    const float B4  = 0.8660254037844386f;      // sqrt(3/4)
    const float A5  = 0.6324555320336759f;      // sqrt(2/5)
    const float B5  = 0.8944271909999159f;      // sqrt(4/5)

    // Prologue: stage chunk 0 into LDS buffer 0.
    if (EXACT || q < total4)
        async_load_f4_to_lds((unsigned)(uintptr_t)&buf[0][tid], x4 + q);

    #pragma unroll
    for (int i = 0; i < CHUNKS_PER_THREAD; ++i) {
        const int qn = q + stride;
        const bool has_next =
            (i + 1 < CHUNKS_PER_THREAD) && (EXACT || qn < total4);

        // Issue next chunk into the other buffer, then wait for current chunk.
        if (has_next)
            async_load_f4_to_lds((unsigned)(uintptr_t)&buf[(i + 1) & 1][tid],
                                 x4 + qn);
        wait_async_le(has_next ? 1 : 0);

        if (EXACT || q < total4) {
            const v4f xv = buf[i & 1][tid];   // ds_load_b128

            float ps[6][4];
            #pragma unroll
            for (int j = 0; j < 4; ++j) {
                const float t  = fast_tanh(xv[j]) * S;
                const float g  = __builtin_amdgcn_exp2f(t * t * NE); // v_exp_f32
                const float p0 = C0 * g;
                const float p1 = C1 * t * g;
                const float p2 = fmaf(t,      p1, -B2 * p0);
                const float p3 = fmaf(A3 * t, p2, -B3 * p1);
                const float p4 = fmaf(A4 * t, p3, -B4 * p2);
                const float p5 = fmaf(A5 * t, p4, -B5 * p3);
                ps[0][j] = p0; ps[1][j] = p1; ps[2][j] = p2;
                ps[3][j] = p3; ps[4][j] = p4; ps[5][j] = p5;
            }

            // Output layout [B, n, R, N]: row = q / (4096/4); 6 rows of 1024
            // float4 each per input row. Write-once stream -> non-temporal.
            const int row  = q >> 10;          // 4096 / 4 = 1024 float4 / row
            const int col  = q & 1023;
            v4f* ob = out4 + (size_t)row * (6 * 1024) + col;
            #pragma unroll
            for (int k = 0; k < 6; ++k) {
                v4f o = { ps[k][0], ps[k][1], ps[k][2], ps[k][3] };
                __builtin_nontemporal_store(o, ob + (size_t)k * 1024);
            }
        }
        q = qn;
    }
}

extern "C" void kernel_launch(void* const* d_in, const int* in_sizes, int n_in,
                              void* d_out, int out_size, void* d_ws, size_t ws_size,
                              hipStream_t stream) {
    (void)n_in; (void)out_size; (void)d_ws; (void)ws_size;
    const v4f* x4   = (const v4f*)d_in[0];
    v4f*       out4 = (v4f*)d_out;

    const int total  = in_sizes[0];       // 32*64*4096 = 8,388,608 floats
    const int total4 = total >> 2;        // 2,097,152 float4 chunks

    const int per_block = BLOCK * CHUNKS_PER_THREAD;             // 2048 chunks
    const int blocks    = (total4 + per_block - 1) / per_block;  // 1024

    if (total4 % per_block == 0) {
        // Exact coverage (reference shape): guard-free straight-line pipeline.
        HermiteFuncs_kernel<true><<<blocks, BLOCK, 0, stream>>>(x4, out4, total4);
    } else {
        HermiteFuncs_kernel<false><<<blocks, BLOCK, 0, stream>>>(x4, out4, total4);
    }
}